// GraphSAGESummarizer_31456340476253
// MI455X (gfx1250) — compile-verified
//
#include <hip/hip_runtime.h>
#include <hip/hip_bf16.h>
#include <math.h>

#define N_NODES 50000
#define N_EDGES 600000
#define CH      128
#define N_GRAPHS 64
#define SCAN_T  1024

typedef __attribute__((ext_vector_type(2))) float v2f;
typedef __attribute__((ext_vector_type(8))) float v8f;

// ---------------------------------------------------------------- zero (int)
__global__ void zero_int_kernel(int* __restrict__ p, int n) {
    int i = blockIdx.x * blockDim.x + threadIdx.x;
    if (i < n) p[i] = 0;
}

__global__ void zero_f_kernel(float* __restrict__ p, long n) {
    long i = (long)blockIdx.x * blockDim.x + threadIdx.x;
    long stride = (long)gridDim.x * blockDim.x;
    for (; i < n; i += stride) p[i] = 0.0f;
}

// ---------------------------------------------------------------- degree count (one-time)
__global__ void count_kernel(const int* __restrict__ dst, int* __restrict__ deg) {
    int e = blockIdx.x * blockDim.x + threadIdx.x;
    if (e < N_EDGES) atomicAdd(&deg[dst[e]], 1);
}

// ---------------------------------------------------------------- exclusive prefix scan over 50000
// degrees, single workgroup, LDS partials (one-time). rowstart has N_NODES+1 entries.
__global__ void __launch_bounds__(SCAN_T)
scan_kernel(const int* __restrict__ deg, int* __restrict__ rowstart) {
    __shared__ int part[SCAN_T];
    const int t = threadIdx.x;
    const int chunk = (N_NODES + SCAN_T - 1) / SCAN_T;   // 49
    const int lo = t * chunk;
    const int hi = (lo + chunk < N_NODES) ? lo + chunk : N_NODES;
    int s = 0;
    for (int i = lo; i < hi; ++i) s += deg[i];
    part[t] = s;
    __syncthreads();
    // Hillis-Steele inclusive scan over the 1024 partials
    for (int off = 1; off < SCAN_T; off <<= 1) {
        int v = (t >= off) ? part[t - off] : 0;
        __syncthreads();
        part[t] += v;
        __syncthreads();
    }
    int run = (t == 0) ? 0 : part[t - 1];
    for (int i = lo; i < hi; ++i) { rowstart[i] = run; run += deg[i]; }
    if (t == SCAN_T - 1) rowstart[N_NODES] = run;        // == N_EDGES
}

__global__ void copy_int_kernel(const int* __restrict__ a, int* __restrict__ b, int n) {
    int i = blockIdx.x * blockDim.x + threadIdx.x;
    if (i < n) b[i] = a[i];
}

// ---------------------------------------------------------------- CSR fill: counting sort of src by
// dst (one-time, 600K atomics)
__global__ void fill_csr_kernel(const int* __restrict__ src, const int* __restrict__ dst,
                                int* __restrict__ cursor, int* __restrict__ csr_src) {
    int e = blockIdx.x * blockDim.x + threadIdx.x;
    if (e < N_EDGES) {
        int slot = atomicAdd(&cursor[dst[e]], 1);
        csr_src[slot] = src[e];
    }
}

// ---------------------------------------------------------------- mean aggregation as pure GATHER:
// one wave per node; each lane owns a float4 channel slice (32 lanes x 4 = 128 ch).
// Neighbor rows are 512B contiguous -> coalesced L2-resident streams, zero atomics.
__global__ void __launch_bounds__(256)
gather_kernel(const int* __restrict__ rowstart, const int* __restrict__ csr_src,
              const float* __restrict__ x, float* __restrict__ agg) {
    const int node = blockIdx.x * 8 + (threadIdx.x >> 5);   // 6250 * 8 == 50000 exactly
    const int lane = threadIdx.x & 31;
    const int beg = rowstart[node];
    const int end = rowstart[node + 1];
    float4 acc = make_float4(0.f, 0.f, 0.f, 0.f);
    for (int j = beg; j < end; ++j) {
        const int s = csr_src[j];
        const float4 v = *(const float4*)(x + (long)s * CH + lane * 4);
        acc.x += v.x; acc.y += v.y; acc.z += v.z; acc.w += v.w;
    }
    const float inv = 1.0f / fmaxf((float)(end - beg), 1.0f);  // mean (deg==0 -> 0 anyway)
    acc.x *= inv; acc.y *= inv; acc.z *= inv; acc.w *= inv;
    *(float4*)(agg + (long)node * CH + lane * 4) = acc;
}

// ---------------------------------------------------------------- fused SAGE GEMM:
// out = relu( agg_norm @ Wl + bias + x @ Wr ); V_WMMA_F32_16X16X4_F32,
// one wave32 per 16x16 tile, 3125 x 8 tiles exactly, no divergence (EXEC all 1s).
__global__ void __launch_bounds__(128)
sage_gemm_kernel(const float* __restrict__ agg, const float* __restrict__ xin,
                 const float* __restrict__ Wl,  const float* __restrict__ Wr,
                 const float* __restrict__ bias, float* __restrict__ out) {
    const int wave = threadIdx.x >> 5;
    const int lane = threadIdx.x & 31;
    const int half = lane >> 4;               // 0: K={0,1}, 1: K={2,3} slice of 16x4 A frag
    const int lid  = lane & 15;
    const int w    = blockIdx.x * 4 + wave;   // 25000 tiles
    const int tm   = w >> 3;
    const int tn   = w & 7;
    const int row  = tm * 16 + lid;           // A row this lane owns
    const int col  = tn * 16 + lid;           // B/D column this lane owns

    v8f c = {};
#pragma unroll 4
    for (int k0 = 0; k0 < CH; k0 += 4) {
        const int ka = k0 + 2 * half;
        v2f a1 = *(const v2f*)(agg + (long)row * CH + ka);
        v2f a2 = *(const v2f*)(xin + (long)row * CH + ka);
        v2f bl, br;
        bl.x = Wl[(ka + 0) * CH + col];
        bl.y = Wl[(ka + 1) * CH + col];
        br.x = Wr[(ka + 0) * CH + col];
        br.y = Wr[(ka + 1) * CH + col];
        c = __builtin_amdgcn_wmma_f32_16x16x4_f32(false, a1, false, bl, (short)0, c, false, false);
        c = __builtin_amdgcn_wmma_f32_16x16x4_f32(false, a2, false, br, (short)0, c, false, false);
    }

    const float bv  = bias[col];
    const int mbase = tm * 16 + half * 8;     // D: lanes<16 -> M=tm*16+r, lanes>=16 -> +8
#pragma unroll
    for (int r = 0; r < 8; ++r) {
        float v = c[r] + bv;
        out[(long)(mbase + r) * CH + col] = v > 0.0f ? v : 0.0f;
    }
}

// ---------------------------------------------------------------- global mean pool (one pass)
__global__ void pool_kernel(const float* __restrict__ h, const int* __restrict__ batch,
                            float* __restrict__ pooled, float* __restrict__ gcnt) {
    long idx = (long)blockIdx.x * blockDim.x + threadIdx.x;
    if (idx >= (long)N_NODES * 32) return;
    int i  = (int)(idx >> 5);
    int c4 = (int)(idx & 31) << 2;
    int g  = batch[i];
    const float4 v = *(const float4*)(h + (long)i * CH + c4);
    float* p = pooled + g * CH + c4;
    atomicAdd(p + 0, v.x);
    atomicAdd(p + 1, v.y);
    atomicAdd(p + 2, v.z);
    atomicAdd(p + 3, v.w);
    if (c4 == 0) atomicAdd(&gcnt[g], 1.0f);
}

// ---------------------------------------------------------------- head: sigmoid(mean_pool @ Wc + bc)
__global__ void head_kernel(const float* __restrict__ pooled, const float* __restrict__ gcnt,
                            const float* __restrict__ Wc, const float* __restrict__ bc,
                            float* __restrict__ out) {
    int g = blockIdx.x;
    int lane = threadIdx.x;  // 32 threads = 1 wave
    float s = 0.0f;
    for (int c = lane; c < CH; c += 32) s += pooled[g * CH + c] * Wc[c];
#pragma unroll
    for (int off = 16; off > 0; off >>= 1) s += __shfl_xor(s, off, 32);
    if (lane == 0) {
        float z = s / fmaxf(gcnt[g], 1.0f) + bc[0];
        out[g] = 1.0f / (1.0f + __expf(-z));
    }
}

// ---------------------------------------------------------------- launcher
extern "C" void kernel_launch(void* const* d_in, const int* in_sizes, int n_in,
                              void* d_out, int out_size, void* d_ws, size_t ws_size,
                              hipStream_t stream) {
    const float* x     = (const float*)d_in[0];
    const int*   ei    = (const int*)d_in[1];     // [2, E]
    const int*   batch = (const int*)d_in[3];
    const float* Wl[4] = { (const float*)d_in[4],  (const float*)d_in[7],
                           (const float*)d_in[10], (const float*)d_in[13] };
    const float* Wr[4] = { (const float*)d_in[5],  (const float*)d_in[8],
                           (const float*)d_in[11], (const float*)d_in[14] };
    const float* bs[4] = { (const float*)d_in[6],  (const float*)d_in[9],
                           (const float*)d_in[12], (const float*)d_in[15] };
    const float* Wc = (const float*)d_in[16];
    const float* bc = (const float*)d_in[17];

    const int* src = ei;
    const int* dst = ei + N_EDGES;

    // workspace layout
    const long nNC = (long)N_NODES * CH;
    float* agg    = (float*)d_ws;                 // N*CH
    float* hA     = agg + nNC;                    // N*CH
    float* hB     = hA + nNC;                     // N*CH
    float* pooled = hB + nNC;                     // G*CH
    float* gcnt   = pooled + (long)N_GRAPHS * CH; // G
    int*   deg      = (int*)(gcnt + N_GRAPHS);    // N
    int*   rowstart = deg + N_NODES;              // N+1
    int*   cursor   = rowstart + N_NODES + 1;     // N
    int*   csr_src  = cursor + N_NODES;           // E

    // ---- one-time CSR build (deterministic per call) ----
    zero_int_kernel<<<(N_NODES + 255) / 256, 256, 0, stream>>>(deg, N_NODES);
    count_kernel<<<(N_EDGES + 255) / 256, 256, 0, stream>>>(dst, deg);
    scan_kernel<<<1, SCAN_T, 0, stream>>>(deg, rowstart);
    copy_int_kernel<<<(N_NODES + 255) / 256, 256, 0, stream>>>(rowstart, cursor, N_NODES);
    fill_csr_kernel<<<(N_EDGES + 255) / 256, 256, 0, stream>>>(src, dst, cursor, csr_src);

    // ---- 4 SAGE layers: gather (no atomics) + WMMA GEMM ----
    const int gatherBlocks = N_NODES / 8;                      // 6250
    const int gemmBlocks   = (N_NODES / 16) * (CH / 16) / 4;   // 6250

    const float* cur = x;
    float* bufs[2] = { hA, hB };
    for (int l = 0; l < 4; ++l) {
        float* nxt = bufs[l & 1];
        gather_kernel<<<gatherBlocks, 256, 0, stream>>>(rowstart, csr_src, cur, agg);
        sage_gemm_kernel<<<gemmBlocks, 128, 0, stream>>>(agg, cur, Wl[l], Wr[l], bs[l], nxt);
        cur = nxt;
    }

    // ---- pool + head ----
    zero_f_kernel<<<64, 256, 0, stream>>>(pooled, (long)N_GRAPHS * CH + N_GRAPHS);
    const long poolThreads = (long)N_NODES * 32;
    pool_kernel<<<(int)((poolThreads + 255) / 256), 256, 0, stream>>>(cur, batch, pooled, gcnt);
    head_kernel<<<N_GRAPHS, 32, 0, stream>>>(pooled, gcnt, Wc, bc, (float*)d_out);
}